// GCN_46608985096492
// MI455X (gfx1250) — compile-verified
//
#include <hip/hip_runtime.h>
#include <math.h>

#define DIM   128
#define NCLS  10

typedef __attribute__((ext_vector_type(2))) float v2f;
typedef __attribute__((ext_vector_type(8))) float v8f;

// -------------------- degree / normalization --------------------
__global__ void deg_kernel(const int* __restrict__ dst, float* __restrict__ deg, int n_edges) {
    int i = blockIdx.x * blockDim.x + threadIdx.x;
    if (i < n_edges) atomicAdd(&deg[dst[i]], 1.0f);
}

__global__ void dinv_kernel(float* __restrict__ deg, int n_nodes) {
    int i = blockIdx.x * blockDim.x + threadIdx.x;
    if (i < n_nodes) deg[i] = rsqrtf(deg[i] + 1.0f); // +1 self loop; in-place deg -> dinv
}

// -------------------- H = (relu?)X @ W  via V_WMMA_F32_16X16X4_F32 --------------------
// Block = 256 threads = 8 waves. Block computes 16 rows x 128 cols; wave w owns cols [16w,16w+16).
__global__ void __launch_bounds__(256) gemm128(const float* __restrict__ X,
                                               const float* __restrict__ W,
                                               float* __restrict__ H,
                                               int n_rows, int relu_in)
{
    __shared__ float sX[16 * 132];     // X tile, padded stride 132 -> conflict-free A reads
    __shared__ float sW[DIM * DIM];    // full 128x128 W (64 KB; WGP LDS = 320 KB)

    const int t = threadIdx.x;
    const int rowBase = blockIdx.x << 4;

    // stage X tile (16x128), fusing ReLU of the previous layer's output
    for (int i = t; i < 16 * 32; i += 256) {
        int r = i >> 5, c4 = i & 31;
        float4 v = make_float4(0.f, 0.f, 0.f, 0.f);
        if (rowBase + r < n_rows)
            v = ((const float4*)(X + (size_t)(rowBase + r) * DIM))[c4];
        if (relu_in) {
            v.x = fmaxf(v.x, 0.f); v.y = fmaxf(v.y, 0.f);
            v.z = fmaxf(v.z, 0.f); v.w = fmaxf(v.w, 0.f);
        }
        float* p = sX + r * 132 + c4 * 4;
        p[0] = v.x; p[1] = v.y; p[2] = v.z; p[3] = v.w;
    }
    // stage W (128x128)
    for (int i = t; i < DIM * DIM / 4; i += 256)
        ((float4*)sW)[i] = ((const float4*)W)[i];
    __syncthreads();

    const int wave = t >> 5;
    const int lane = t & 31;
    const int half = lane >> 4;     // 0: K=k0,k0+1 ; 1: K=k0+2,k0+3 (ISA 16x4 f32 layout)
    const int l16  = lane & 15;
    const int n0   = wave << 4;

    v8f acc = {};
#pragma unroll
    for (int k0 = 0; k0 < DIM; k0 += 4) {
        const int ka = k0 + (half << 1);
        v2f a, b;
        a.x = sX[l16 * 132 + ka];            // A: lane l16 = row M, K striped per layout
        a.y = sX[l16 * 132 + ka + 1];
        b.x = sW[ka * DIM + n0 + l16];       // B: lane l16 = col N, same K striping
        b.y = sW[(ka + 1) * DIM + n0 + l16];
        acc = __builtin_amdgcn_wmma_f32_16x16x4_f32(
            /*neg_a=*/false, a, /*neg_b=*/false, b,
            /*c_mod=*/(short)0, acc, /*reuse_a=*/false, /*reuse_b=*/false);
    }

    // C/D layout: VGPR i -> M = i + 8*half, N = l16
#pragma unroll
    for (int i = 0; i < 8; ++i) {
        int m = rowBase + i + (half << 3);
        if (m < n_rows) H[(size_t)m * DIM + n0 + l16] = acc[i];
    }
}

// -------------------- out = bias + h * dinv^2  (self-loop term, also initializes out) ----
__global__ void init_out(const float* __restrict__ tmpH, const float* __restrict__ bias,
                         const float* __restrict__ dinv, float* __restrict__ out, int n_nodes)
{
    int i = blockIdx.x * blockDim.x + threadIdx.x;   // one float4 per thread
    int total = n_nodes * (DIM / 4);
    if (i >= total) return;
    int node = i >> 5, c4 = i & 31;
    float dd = dinv[node]; dd *= dd;
    float4 h = ((const float4*)tmpH)[i];
    float4 b = ((const float4*)bias)[c4];
    ((float4*)out)[i] = make_float4(b.x + h.x * dd, b.y + h.y * dd,
                                    b.z + h.z * dd, b.w + h.w * dd);
}

// -------------------- edge scatter: out[d] += h[s] * dinv[s]*dinv[d] --------------------
// One wave per edge; each lane moves one float4 (L2-resident atomics on MI455X's 192MB L2).
__global__ void __launch_bounds__(256) scatter_edges(const float* __restrict__ tmpH,
                                                     const int* __restrict__ src,
                                                     const int* __restrict__ dst,
                                                     const float* __restrict__ dinv,
                                                     float* __restrict__ out, int n_edges)
{
    int e = blockIdx.x * 8 + (threadIdx.x >> 5);
    if (e >= n_edges) return;
    int lane = threadIdx.x & 31;
    int s = src[e], d = dst[e];
    float nrm = dinv[s] * dinv[d];
    float4 v = ((const float4*)(tmpH + (size_t)s * DIM))[lane];
    float* o = out + (size_t)d * DIM + lane * 4;
    atomicAdd(o + 0, v.x * nrm);
    atomicAdd(o + 1, v.y * nrm);
    atomicAdd(o + 2, v.z * nrm);
    atomicAdd(o + 3, v.w * nrm);
}

// -------------------- global mean pool (sums + counts) --------------------
__global__ void __launch_bounds__(256) pool_kernel(const float* __restrict__ h,
                                                   const int* __restrict__ batch,
                                                   float* __restrict__ sums,
                                                   float* __restrict__ cnt, int n_nodes)
{
    int node = blockIdx.x * 8 + (threadIdx.x >> 5);
    if (node >= n_nodes) return;
    int lane = threadIdx.x & 31;
    int g = batch[node];
    float4 v = ((const float4*)(h + (size_t)node * DIM))[lane];
    float* o = sums + (size_t)g * DIM + lane * 4;
    atomicAdd(o + 0, v.x); atomicAdd(o + 1, v.y);
    atomicAdd(o + 2, v.z); atomicAdd(o + 3, v.w);
    if (lane == 0) atomicAdd(&cnt[g], 1.0f);
}

// -------------------- classifier + log_softmax --------------------
__global__ void classifier(const float* __restrict__ sums, const float* __restrict__ cnt,
                           const float* __restrict__ Wl, const float* __restrict__ bl,
                           float* __restrict__ out, int n_graphs)
{
    int g = blockIdx.x * blockDim.x + threadIdx.x;
    if (g >= n_graphs) return;
    float inv = 1.0f / fmaxf(cnt[g], 1.0f);
    float logit[NCLS];
#pragma unroll
    for (int c = 0; c < NCLS; ++c) logit[c] = bl[c];
    for (int k = 0; k < DIM; ++k) {
        float hv = sums[(size_t)g * DIM + k] * inv;
#pragma unroll
        for (int c = 0; c < NCLS; ++c) logit[c] += hv * Wl[k * NCLS + c];
    }
    float m = logit[0];
#pragma unroll
    for (int c = 1; c < NCLS; ++c) m = fmaxf(m, logit[c]);
    float sse = 0.f;
#pragma unroll
    for (int c = 0; c < NCLS; ++c) sse += __expf(logit[c] - m);
    float lse = m + __logf(sse);
#pragma unroll
    for (int c = 0; c < NCLS; ++c) out[g * NCLS + c] = logit[c] - lse;
}

// -------------------- host orchestration --------------------
extern "C" void kernel_launch(void* const* d_in, const int* in_sizes, int n_in,
                              void* d_out, int out_size, void* d_ws, size_t ws_size,
                              hipStream_t stream)
{
    const float* x   = (const float*)d_in[0];
    const int* eidx  = (const int*)d_in[1];
    const int* batch = (const int*)d_in[2];
    const float* W1 = (const float*)d_in[3]; const float* b1 = (const float*)d_in[4];
    const float* W2 = (const float*)d_in[5]; const float* b2 = (const float*)d_in[6];
    const float* W3 = (const float*)d_in[7]; const float* b3 = (const float*)d_in[8];
    const float* Wl = (const float*)d_in[9]; const float* bl = (const float*)d_in[10];
    float* out = (float*)d_out;

    const int n_nodes  = in_sizes[0] / DIM;
    const int n_edges  = in_sizes[1] / 2;
    const int n_graphs = out_size / NCLS;
    const int* src = eidx;
    const int* dst = eidx + n_edges;

    // workspace partition (all node feature buffers ~154MB: fits MI455X's 192MB L2)
    char* p = (char*)d_ws;
    const size_t node_bytes = (((size_t)n_nodes * DIM * sizeof(float)) + 255) & ~(size_t)255;
    float* A    = (float*)p; p += node_bytes;              // tmp: X@W per layer
    float* B    = (float*)p; p += node_bytes;              // layer1 / layer3 output
    float* C    = (float*)p; p += node_bytes;              // layer2 output
    float* dinv = (float*)p; p += ((size_t)n_nodes * 4 + 255) & ~(size_t)255;
    float* sums = (float*)p; p += ((size_t)n_graphs * DIM * 4 + 255) & ~(size_t)255;
    float* cnt  = (float*)p;

    hipMemsetAsync(dinv, 0, (size_t)n_nodes * sizeof(float), stream);
    hipMemsetAsync(sums, 0, (size_t)n_graphs * DIM * sizeof(float), stream);
    hipMemsetAsync(cnt,  0, (size_t)n_graphs * sizeof(float), stream);

    deg_kernel <<<(n_edges + 255) / 256, 256, 0, stream>>>(dst, dinv, n_edges);
    dinv_kernel<<<(n_nodes + 255) / 256, 256, 0, stream>>>(dinv, n_nodes);

    const int tiles  = (n_nodes + 15) / 16;
    const int e4grid = (n_nodes * 32 + 255) / 256;
    const int eblk   = (n_edges + 7) / 8;

    // layer 1: in = x (no relu) -> B
    gemm128      <<<tiles, 256, 0, stream>>>(x, W1, A, n_nodes, 0);
    init_out     <<<e4grid, 256, 0, stream>>>(A, b1, dinv, B, n_nodes);
    scatter_edges<<<eblk,  256, 0, stream>>>(A, src, dst, dinv, B, n_edges);
    // layer 2: in = relu(B) -> C
    gemm128      <<<tiles, 256, 0, stream>>>(B, W2, A, n_nodes, 1);
    init_out     <<<e4grid, 256, 0, stream>>>(A, b2, dinv, C, n_nodes);
    scatter_edges<<<eblk,  256, 0, stream>>>(A, src, dst, dinv, C, n_edges);
    // layer 3: in = relu(C) -> B (no relu on output)
    gemm128      <<<tiles, 256, 0, stream>>>(C, W3, A, n_nodes, 1);
    init_out     <<<e4grid, 256, 0, stream>>>(A, b3, dinv, B, n_nodes);
    scatter_edges<<<eblk,  256, 0, stream>>>(A, src, dst, dinv, B, n_edges);

    // pool + classify + log_softmax
    pool_kernel<<<(n_nodes + 7) / 8, 256, 0, stream>>>(B, batch, sums, cnt, n_nodes);
    classifier <<<(n_graphs + 127) / 128, 128, 0, stream>>>(sums, cnt, Wl, bl, out, n_graphs);
}